// AMLGCN_3822520893440
// MI455X (gfx1250) — compile-verified
//
#include <hip/hip_runtime.h>
#include <hip/hip_bf16.h>

typedef __attribute__((ext_vector_type(2))) float v2f;
typedef __attribute__((ext_vector_type(8))) float v8f;

// ---------------------------------------------------------------------------
// h[M,N] = (RELU_A ? relu(A) : A)[M,K] @ B[K,N]   (all f32, row-major)
// One wave32 per 16x16 output tile; K-loop in steps of 4 using
// V_WMMA_F32_16X16X4_F32 (full f32 precision, matches f32 reference).
//
// A operand layout (32-bit A 16x4): lane l holds row M=l%16,
//   VGPR j holds K = 2*(l/16) + j.
// B operand layout (32-bit B 4x16): lane l holds col N=l%16,
//   VGPR j holds K = 2*(l/16) + j.
// C/D (16x16 f32, 8 VGPRs): VGPR r, lane l -> M = r + 8*(l/16), N = l%16.
// ---------------------------------------------------------------------------
template <bool RELU_A>
__global__ __launch_bounds__(32)
void gcn_gemm_wmma(const float* __restrict__ A, const float* __restrict__ B,
                   float* __restrict__ C, int M, int K, int N) {
  const int tm   = blockIdx.x;        // M/16 tile
  const int tn   = blockIdx.y;        // N/16 tile
  const int lane = threadIdx.x;       // 0..31
  const int mrow = lane & 15;
  const int half = lane >> 4;         // 0 or 1
  const int kgrp = half * 2;          // K base within a 4-step
  const int ncol = tn * 16 + (lane & 15);

  const float* __restrict__ Arow = A + (size_t)(tm * 16 + mrow) * K;

  v8f acc = {};
  for (int k0 = 0; k0 < K; k0 += 4) {
    float a0 = Arow[k0 + kgrp + 0];
    float a1 = Arow[k0 + kgrp + 1];
    if (RELU_A) {
      a0 = fmaxf(a0, 0.0f);
      a1 = fmaxf(a1, 0.0f);
    }
    v2f a; a.x = a0; a.y = a1;
    v2f b;
    b.x = B[(size_t)(k0 + kgrp + 0) * N + ncol];
    b.y = B[(size_t)(k0 + kgrp + 1) * N + ncol];
    acc = __builtin_amdgcn_wmma_f32_16x16x4_f32(
        /*neg_a=*/false, a, /*neg_b=*/false, b,
        /*c_mod=*/(short)0, acc, /*reuse_a=*/false, /*reuse_b=*/false);
  }

  const int crow = tm * 16 + half * 8;
#pragma unroll
  for (int r = 0; r < 8; ++r) {
    C[(size_t)(crow + r) * N + ncol] = acc[r];
  }
}

// ---------------------------------------------------------------------------
// agg[n, f] = bias[f]  (fold "+b" into the scatter target)
// ---------------------------------------------------------------------------
__global__ void gcn_init_bias(float* __restrict__ agg, const float* __restrict__ b,
                              long long total, int Fmask) {
  long long stride = (long long)gridDim.x * blockDim.x;
  for (long long i = (long long)blockIdx.x * blockDim.x + threadIdx.x; i < total;
       i += stride) {
    agg[i] = b[(int)i & Fmask];   // F is a power of two (128 / 64)
  }
}

// ---------------------------------------------------------------------------
// Edge scatter: agg[dst] += ew[e] * h[src]
// One thread per (edge, float4-chunk); chunk index fastest so a wave reads one
// (or two) contiguous source rows -> coalesced gather that lives in the
// 192MB L2. Scatter via global_atomic_add_f32.
// ---------------------------------------------------------------------------
__global__ void gcn_scatter(const float* __restrict__ h, const int* __restrict__ src,
                            const int* __restrict__ dst, const float* __restrict__ ew,
                            float* __restrict__ agg, long long E, int F) {
  const int chunks = F >> 2;                       // float4 chunks per row
  const long long total = E * chunks;
  const long long stride = (long long)gridDim.x * blockDim.x;
  for (long long i = (long long)blockIdx.x * blockDim.x + threadIdx.x; i < total;
       i += stride) {
    const int e = (int)(i / chunks);
    const int c = (int)(i - (long long)e * chunks);
    const int s = src[e];
    const int d = dst[e];
    const float w = ew[e];
    const float4 hv = *(const float4*)(h + (size_t)s * F + (size_t)c * 4);
    float* ap = agg + (size_t)d * F + (size_t)c * 4;
    atomicAdd(ap + 0, w * hv.x);
    atomicAdd(ap + 1, w * hv.y);
    atomicAdd(ap + 2, w * hv.z);
    atomicAdd(ap + 3, w * hv.w);
  }
}

// ---------------------------------------------------------------------------
// out[n, 0:2] = relu(agg2[n, 0:64]) @ Wl[64,2] + bl[0:2]
// ---------------------------------------------------------------------------
__global__ void gcn_final(const float* __restrict__ agg2, const float* __restrict__ Wl,
                          const float* __restrict__ bl, float* __restrict__ out, int n) {
  const int i = blockIdx.x * blockDim.x + threadIdx.x;
  if (i >= n) return;
  const float* a = agg2 + (size_t)i * 64;
  float o0 = bl[0], o1 = bl[1];
#pragma unroll 8
  for (int k = 0; k < 64; ++k) {
    const float v = fmaxf(a[k], 0.0f);
    o0 = fmaf(v, Wl[k * 2 + 0], o0);
    o1 = fmaf(v, Wl[k * 2 + 1], o1);
  }
  out[(size_t)i * 2 + 0] = o0;
  out[(size_t)i * 2 + 1] = o1;
}

extern "C" void kernel_launch(void* const* d_in, const int* in_sizes, int n_in,
                              void* d_out, int out_size, void* d_ws, size_t ws_size,
                              hipStream_t stream) {
  const int IN_C = 128, HID = 128, H2 = 64;
  const float* x  = (const float*)d_in[0];
  const int*   ei = (const int*)d_in[1];       // shape (2, E): src = ei, dst = ei + E
  const float* ew = (const float*)d_in[2];
  const float* W1 = (const float*)d_in[3];
  const float* b1 = (const float*)d_in[4];
  const float* W2 = (const float*)d_in[5];
  const float* b2 = (const float*)d_in[6];
  const float* Wl = (const float*)d_in[7];
  const float* bl = (const float*)d_in[8];
  float* out = (float*)d_out;

  const int N = in_sizes[0] / IN_C;            // 50000 (= 3125 * 16)
  const long long E = (long long)in_sizes[2];  // 800000
  const int* src = ei;
  const int* dst = ei + E;

  // Workspace ping-pong: [h buffer | agg buffer], each N*HID floats.
  float* h_buf   = (float*)d_ws;
  float* agg_buf = h_buf + (size_t)N * HID;

  const int mt = N / 16;                       // 3125 M-tiles

  // ---- Layer 1: agg1 = b1; h1 = x@W1; agg1[dst] += ew * h1[src] ----
  {
    long long tot = (long long)N * HID;
    int blocks = (int)((tot + 255) / 256);
    gcn_init_bias<<<blocks, 256, 0, stream>>>(agg_buf, b1, tot, HID - 1);
  }
  gcn_gemm_wmma<false><<<dim3(mt, HID / 16), 32, 0, stream>>>(x, W1, h_buf, N, IN_C, HID);
  {
    long long tot = E * (HID / 4);
    int blocks = (int)((tot + 255) / 256);
    if (blocks > 131072) blocks = 131072;
    gcn_scatter<<<blocks, 256, 0, stream>>>(h_buf, src, dst, ew, agg_buf, E, HID);
  }

  // ---- Layer 2: h2 = relu(agg1)@W2 (into h_buf); agg2 = b2 (into agg_buf);
  //               agg2[dst] += ew * h2[src] ----
  gcn_gemm_wmma<true><<<dim3(mt, H2 / 16), 32, 0, stream>>>(agg_buf, W2, h_buf, N, HID, H2);
  {
    long long tot = (long long)N * H2;
    int blocks = (int)((tot + 255) / 256);
    gcn_init_bias<<<blocks, 256, 0, stream>>>(agg_buf, b2, tot, H2 - 1);
  }
  {
    long long tot = E * (H2 / 4);
    int blocks = (int)((tot + 255) / 256);
    if (blocks > 131072) blocks = 131072;
    gcn_scatter<<<blocks, 256, 0, stream>>>(h_buf, src, dst, ew, agg_buf, E, H2);
  }

  // ---- Final: out = relu(agg2) @ Wl + bl ----
  gcn_final<<<(N + 255) / 256, 256, 0, stream>>>(agg_buf, Wl, bl, out, N);
}